// Decoder_2018634629437
// MI455X (gfx1250) — compile-verified
//
#include <hip/hip_runtime.h>
#include <math.h>

typedef __attribute__((ext_vector_type(2))) float v2f;
typedef __attribute__((ext_vector_type(8))) float v8f;

#define HH 1024
#define LL 512
#define VV 50257

// workspace layout (float offsets)
#define WS_EH   0        // concat(e, h0)            [2H]
#define WS_ECTX 2048     // concat(e, ctx)           [2H]
#define WS_ATTN 4096     // attn logits -> weights   [L]
#define WS_XG   4608     // relu(comb)               [H]
#define WS_GI   5632     // w_ih @ xg + b_ih         [3H]
#define WS_GH   8704     // w_hh @ h0 + b_hh         [3H]
#define WS_H1   11776    // new hidden               [H]

// ---------------------------------------------------------------------------
// Stage 1: embedding lookup + build concat vectors
__global__ void prep_kernel(const int* __restrict__ x, const float* __restrict__ hidden,
                            const float* __restrict__ emb, float* __restrict__ ws) {
    int tid = blockIdx.x * blockDim.x + threadIdx.x;   // 0..2047
    if (tid < HH) {
        float e = emb[(size_t)x[0] * HH + tid];
        ws[WS_EH + tid]   = e;   // concat(e, h0) low half
        ws[WS_ECTX + tid] = e;   // concat(e, ctx) low half
    } else if (tid < 2 * HH) {
        ws[WS_EH + tid] = hidden[tid - HH];
    }
}

// ---------------------------------------------------------------------------
// Generic fp32 GEMV  y = act(W @ v + b)  using V_WMMA_F32_16X16X4_F32.
// W is [M,K] row-major. One wave computes 16 output rows.
//
// K is processed 8 columns per iteration with a permuted K mapping (dot
// products are order-independent; we build both A and B with the SAME
// mapping, so the result is exact):
//   lane-lo (0-15)  loads float4 W[row][k..k+3]     (one b128 per lane)
//   lane-hi (16-31) loads float4 W[row][k+4..k+7]
//   WMMA#0 uses slots {0,1}: hw-K 0,1 = cols k,k+1 (lo); hw-K 2,3 = k+4,k+5 (hi)
//   WMMA#1 uses slots {2,3}: hw-K 0,1 = cols k+2,k+3;    hw-K 2,3 = k+6,k+7
// B tile: only column N=0 nonzero -> lane 0 / lane 16 carry the matching v[]
// pairs; all other lanes zeroed via multiply mask (EXEC stays all-1s).
// D column N=0: lane 0 vgprs 0-7 = rows 0-7, lane 16 vgprs 0-7 = rows 8-15.
__launch_bounds__(256)
__global__ void gemv_wmma_kernel(const float* __restrict__ W, const float* __restrict__ vin,
                                 const float* __restrict__ bias, float* __restrict__ y,
                                 int M, int K, int relu) {
    __shared__ __align__(16) float sv[2 * HH];
    for (int i = threadIdx.x; i < K; i += 256) sv[i] = vin[i];
    __syncthreads();

    const int wave = threadIdx.x >> 5;
    const int lane = threadIdx.x & 31;
    const int tile = blockIdx.x * 8 + wave;
    const int row0 = tile * 16;
    if (row0 >= M) return;                      // wave-uniform early exit

    const int hi   = lane >> 4;                 // 0: cols k..k+3, 1: cols k+4..k+7
    const int mrow = lane & 15;
    int r = row0 + mrow;
    if (r > M - 1) r = M - 1;                   // clamp OOB rows (keeps EXEC full)
    const float* wrow = W + (size_t)r * K;
    const float msk = (mrow == 0) ? 1.0f : 0.0f;

    v8f acc = {};
    #pragma unroll 4
    for (int k = 0; k < K; k += 8) {
        const int kb = k + (hi << 2);
        const float4 wa = *(const float4*)(wrow + kb);   // global_load_b128
        const float4 va = *(const float4*)(sv + kb);     // ds_load_b128
        v2f a0; a0.x = wa.x;       a0.y = wa.y;
        v2f b0; b0.x = va.x * msk; b0.y = va.y * msk;
        acc = __builtin_amdgcn_wmma_f32_16x16x4_f32(
                  false, a0, false, b0, (short)0, acc, false, false);
        v2f a1; a1.x = wa.z;       a1.y = wa.w;
        v2f b1; b1.x = va.z * msk; b1.y = va.w * msk;
        acc = __builtin_amdgcn_wmma_f32_16x16x4_f32(
                  false, a1, false, b1, (short)0, acc, false, false);
    }

    if (mrow == 0) {                            // lanes 0 and 16 hold column N=0
        const int rb = row0 + (hi << 3);
        #pragma unroll
        for (int i = 0; i < 8; ++i) {
            const int row = rb + i;
            if (row < M) {
                float val = acc[i] + bias[row];
                if (relu) val = fmaxf(val, 0.0f);
                y[row] = val;
            }
        }
    }
}

// ---------------------------------------------------------------------------
// Stage 3: softmax over L=512 logits (single block), write weights to ws + d_out
__global__ void attn_softmax_kernel(float* __restrict__ ws, float* __restrict__ out_attn) {
    __shared__ float red[LL];
    const int t = threadIdx.x;                  // 512 threads
    const float logit = ws[WS_ATTN + t];
    red[t] = logit;
    __syncthreads();
    for (int s = LL / 2; s > 0; s >>= 1) {
        if (t < s) red[t] = fmaxf(red[t], red[t + s]);
        __syncthreads();
    }
    const float mx = red[0];
    __syncthreads();
    const float e = expf(logit - mx);
    red[t] = e;
    __syncthreads();
    for (int s = LL / 2; s > 0; s >>= 1) {
        if (t < s) red[t] += red[t + s];
        __syncthreads();
    }
    const float w = e / red[0];
    ws[WS_ATTN + t] = w;
    out_attn[t] = w;
}

// ---------------------------------------------------------------------------
// Stage 4: ctx[j] = sum_l attn[l] * enc[l, j]  (column reduction, coalesced)
__global__ void ctx_kernel(const float* __restrict__ enc, float* __restrict__ ws) {
    __shared__ float sa[LL];
    for (int i = threadIdx.x; i < LL; i += blockDim.x) sa[i] = ws[WS_ATTN + i];
    __syncthreads();
    const int j = blockIdx.x * blockDim.x + threadIdx.x;   // 0..1023
    float acc = 0.0f;
    #pragma unroll 4
    for (int l = 0; l < LL; ++l)
        acc = fmaf(sa[l], enc[(size_t)l * HH + j], acc);
    ws[WS_ECTX + HH + j] = acc;                 // concat(e, ctx) high half
}

// ---------------------------------------------------------------------------
// Stage 7: GRU gates -> h1
__global__ void gru_kernel(const float* __restrict__ hidden, float* __restrict__ ws,
                           float* __restrict__ out_h1) {
    const int i = blockIdx.x * blockDim.x + threadIdx.x;   // 0..1023
    const float gi_r = ws[WS_GI + i];
    const float gi_z = ws[WS_GI + HH + i];
    const float gi_n = ws[WS_GI + 2 * HH + i];
    const float gh_r = ws[WS_GH + i];
    const float gh_z = ws[WS_GH + HH + i];
    const float gh_n = ws[WS_GH + 2 * HH + i];
    const float r = 1.0f / (1.0f + expf(-(gi_r + gh_r)));
    const float z = 1.0f / (1.0f + expf(-(gi_z + gh_z)));
    const float n = tanhf(gi_n + r * gh_n);
    const float h0 = hidden[i];
    const float h1 = (1.0f - z) * n + z * h0;
    ws[WS_H1 + i] = h1;
    out_h1[i] = h1;
}

// ---------------------------------------------------------------------------
extern "C" void kernel_launch(void* const* d_in, const int* in_sizes, int n_in,
                              void* d_out, int out_size, void* d_ws, size_t ws_size,
                              hipStream_t stream) {
    const int*   x      = (const int*)  d_in[0];
    const float* hidden = (const float*)d_in[1];   // [1,1,H]
    const float* enc    = (const float*)d_in[2];   // [L,H]
    const float* emb    = (const float*)d_in[3];   // [V,H]
    const float* attn_w = (const float*)d_in[4];   // [L,2H]
    const float* attn_b = (const float*)d_in[5];   // [L]
    const float* comb_w = (const float*)d_in[6];   // [H,2H]
    const float* comb_b = (const float*)d_in[7];   // [H]
    const float* w_ih   = (const float*)d_in[8];   // [3H,H]
    const float* w_hh   = (const float*)d_in[9];   // [3H,H]
    const float* b_ih   = (const float*)d_in[10];  // [3H]
    const float* b_hh   = (const float*)d_in[11];  // [3H]
    const float* out_w  = (const float*)d_in[12];  // [V,H]
    const float* out_b  = (const float*)d_in[13];  // [V]
    float* out = (float*)d_out;                    // [V] out | [H] h1 | [L] attn
    float* ws  = (float*)d_ws;

    // e = emb[x]; concat vectors
    prep_kernel<<<8, 256, 0, stream>>>(x, hidden, emb, ws);
    // attn logits = attn_w @ concat(e,h0) + attn_b   (M=512, K=2048)
    gemv_wmma_kernel<<<4, 256, 0, stream>>>(attn_w, ws + WS_EH, attn_b,
                                            ws + WS_ATTN, LL, 2 * HH, 0);
    // softmax -> ws_attn and d_out attn section
    attn_softmax_kernel<<<1, LL, 0, stream>>>(ws, out + VV + HH);
    // ctx = attn @ encoder_outs
    ctx_kernel<<<4, 256, 0, stream>>>(enc, ws);
    // xg = relu(comb_w @ concat(e,ctx) + comb_b)     (M=1024, K=2048)
    gemv_wmma_kernel<<<8, 256, 0, stream>>>(comb_w, ws + WS_ECTX, comb_b,
                                            ws + WS_XG, HH, 2 * HH, 1);
    // gi = w_ih @ xg + b_ih ; gh = w_hh @ h0 + b_hh  (M=3072, K=1024)
    gemv_wmma_kernel<<<24, 256, 0, stream>>>(w_ih, ws + WS_XG, b_ih,
                                             ws + WS_GI, 3 * HH, HH, 0);
    gemv_wmma_kernel<<<24, 256, 0, stream>>>(w_hh, hidden, b_hh,
                                             ws + WS_GH, 3 * HH, HH, 0);
    // GRU gates -> h1 (ws + d_out h1 section)
    gru_kernel<<<4, 256, 0, stream>>>(hidden, ws, out + VV);
    // out = out_w @ h1 + out_b  (M=50257, K=1024)  -- the 206 MB stream
    const int tiles  = (VV + 15) / 16;           // 3142
    const int blocks = (tiles + 7) / 8;          // 393
    gemv_wmma_kernel<<<blocks, 256, 0, stream>>>(out_w, ws + WS_H1, out_b,
                                                 out, VV, HH, 0);
}